// ScatLayerj2_18202071400454
// MI455X (gfx1250) — compile-verified
//
#include <hip/hip_runtime.h>
#include <math.h>

// CDNA5 / gfx1250 fused DTCWT scattering network.
// Column filters are banded 16x16 matmuls on V_WMMA_F32_16X16X4_F32.
// A-band fragments are precomputed into LDS (float2 per lane per k-step) and
// B fragments use a pair-interleaved LDS layout so every fragment is a single
// ds_load_b64 -> straight-line load/WMMA stream, no predication.

typedef float v2f __attribute__((ext_vector_type(2)));
typedef float v8f __attribute__((ext_vector_type(8)));

#define INV_SQRT2F 0.70710678118654752440f
#define MAGBIAS    0.01f
#define MAGBIAS2   1.0e-4f

__device__ __forceinline__ int reflect_idx(int i, int n) {
  i = (i < 0) ? (-1 - i) : i;
  i = (i >= n) ? (2 * n - 1 - i) : i;
  return i;
}

__device__ __forceinline__ float smooth_mag(float re, float im) {
  return sqrtf(re * re + im * im + MAGBIAS2) - MAGBIAS;
}

__device__ __forceinline__ v8f wmma16x16x4(v2f a, v2f b, v8f c) {
  // 8-arg form: (neg_a, A, neg_b, B, c_mod, C, reuse_a, reuse_b)
  return __builtin_amdgcn_wmma_f32_16x16x4_f32(false, a, false, b, (short)0, c,
                                               false, false);
}

// Branchless band tap: h[t] if 0 <= t < taps else 0 (no exec predication).
__device__ __forceinline__ float band_tap(const float* h, int t, int taps) {
  int tc = min(max(t, 0), taps - 1);
  float v = h[tc];
  return (t == tc) ? v : 0.f;
}

// ---------------------------------------------------------------------------
// Level-1 style kernel (fwd_j1 + q2c + smooth magnitude), used twice:
//   MODE 0: first order on x (W=512, C=3): full ll -> out_ll, 6 oriented
//           magnitudes -> out_mag (= p buffer, (b,18,256,256))
//   MODE 1: second order on p (W=256, C=18): avgpool(ll) and the 36
//           second-order magnitudes straight into Z (out_mag = Z).
// One block = one (b, ch, 32x32 tile). 256 threads = 8 waves.
// ---------------------------------------------------------------------------
template <int MODE>
__global__ __launch_bounds__(256) void scat_level1_kernel(
    const float* __restrict__ X, int C, int W,
    const float* __restrict__ g_h0, const float* __restrict__ g_h1,
    float* __restrict__ out_ll, float* __restrict__ out_mag) {
  __shared__ __align__(16) float shf[12];          // h0o (5) | h1o (7)
  __shared__ __align__(16) float aTab[2 * 6 * 64]; // A frags: [f][s][lane]{x,y}
  __shared__ __align__(16) float sx[38 * 40];      // padded input tile
  __shared__ __align__(16) float rbuf[2][20 * 66]; // lo/hi pair-interleaved
  __shared__ __align__(16) float obuf[4][32 * 33]; // ll, lh, hl, hh

  const int tid = threadIdx.x;
  const int nT = W >> 5;
  const int tpc = nT * nT;
  int id = blockIdx.x;
  const int tile = id % tpc;
  id /= tpc;
  const int c_ = id % C;
  const int b = id / C;
  const int tx = tile % nT, ty = tile / nT;
  const int y0 = ty * 32, x0 = tx * 32;

  const float* __restrict__ Xc = X + (size_t)(b * C + c_) * W * W;

  if (tid < 12) shf[tid] = (tid < 5) ? g_h0[tid] : g_h1[tid - 5];
  __syncthreads();  // shf ready for aTab setup

  // -- precompute A-band fragments: A[r][k] = h[k - r - (3-m)], kbase = 16i --
  for (int idx = tid; idx < 2 * 6 * 32; idx += 256) {
    int f = idx / 192;              // 0: h0o (taps 5, m=2), 1: h1o (taps 7, m=3)
    int rem = idx - f * 192;
    int s = rem >> 5, lane = rem & 31;
    int mrow = lane & 15, khalf = (lane >> 4) << 1;
    int k0 = 4 * s + khalf;
    const float* h = f ? (shf + 5) : shf;
    const int taps = f ? 7 : 5;
    const int off = f ? 0 : 1;      // 3 - m
    int t0 = k0 - mrow - off;
    aTab[idx * 2 + 0] = band_tap(h, t0, taps);
    aTab[idx * 2 + 1] = band_tap(h, t0 + 1, taps);
  }

  // -- load symmetric-padded 38x38 input tile ------------------------------
  for (int idx = tid; idx < 38 * 38; idx += 256) {
    int r = idx / 38, cx = idx - r * 38;
    int gy = reflect_idx(y0 - 3 + r, W);
    int gx = reflect_idx(x0 - 3 + cx, W);
    sx[r * 40 + cx] = Xc[(size_t)gy * W + gx];
  }
  __syncthreads();

  // -- row filters (VALU), pair-interleaved store: rbuf[pair*66 + c*2 + (r&1)]
  for (int idx = tid; idx < 40 * 32; idx += 256) {
    int r = idx >> 5, c = idx & 31;
    float lo = 0.f, hi = 0.f;
    if (r < 38) {
      const float* row = &sx[r * 40];
#pragma unroll
      for (int t = 0; t < 5; ++t) lo += row[c + 1 + t] * shf[t];
#pragma unroll
      for (int t = 0; t < 7; ++t) hi += row[c + t] * shf[5 + t];
    }
    const int a = (r >> 1) * 66 + c * 2 + (r & 1);
    rbuf[0][a] = lo;  // rows 38..39 zero (band overrun pad)
    rbuf[1][a] = hi;
  }
  __syncthreads();

  // -- column filters as banded matmuls on WMMA f32 16x16x4 ----------------
  const int lane = tid & 31;
  const int wave = tid >> 5;
  const int mrow = lane & 15;
  const int lhi = lane >> 4;  // 0: K 4s+0/1, 1: K 4s+2/3
#pragma unroll
  for (int rep = 0; rep < 2; ++rep) {
    const int it = wave + rep * 8;
    const int f = it >> 2;            // 0:ll 1:lh 2:hl 3:hh
    const int i = (it >> 1) & 1;      // 16-row sub-tile
    const int j = it & 1;             // 16-col sub-tile
    const float* src = rbuf[f >> 1];  // ll/lh from lo; hl/hh from hi
    const float* at = aTab + (f & 1) * (6 * 64);
    const int bcol = (16 * j + mrow) * 2;
    v8f acc = {0.f, 0.f, 0.f, 0.f, 0.f, 0.f, 0.f, 0.f};
#pragma unroll
    for (int s = 0; s < 6; ++s) {
      v2f a = *(const v2f*)&at[(s * 32 + lane) * 2];
      v2f bv = *(const v2f*)&src[(8 * i + 2 * s + lhi) * 66 + bcol];
      acc = wmma16x16x4(a, bv, acc);
    }
    const int rhalf = lhi << 3;  // D: lanes 16-31 hold rows M+8
#pragma unroll
    for (int v = 0; v < 8; ++v)
      obuf[f][(16 * i + v + rhalf) * 33 + 16 * j + mrow] = acc[v];
  }
  __syncthreads();

  // -- outputs --------------------------------------------------------------
  if (MODE == 0) {
    for (int idx = tid; idx < 1024; idx += 256) {
      int yy = idx >> 5, xx = idx & 31;
      out_ll[((size_t)(b * C + c_) * W + (y0 + yy)) * W + (x0 + xx)] =
          obuf[0][yy * 33 + xx];
    }
  }

  {
    const int u = tid >> 4, v = tid & 15;  // 16x16 subband tile, 1 px/thread
    const int Hs = W >> 1;
    const int ys = (y0 >> 1) + u, xs = (x0 >> 1) + v;
    const int o1 = c_ / 3, ch3 = c_ - 3 * o1;  // only used in MODE 1

    if (MODE == 1) {  // avgpool2 of ll -> Z channels k = 1 + o1
      float s = 0.25f * (obuf[0][(2 * u) * 33 + 2 * v] +
                         obuf[0][(2 * u) * 33 + 2 * v + 1] +
                         obuf[0][(2 * u + 1) * 33 + 2 * v] +
                         obuf[0][(2 * u + 1) * 33 + 2 * v + 1]);
      out_mag[((size_t)(b * 147 + (1 + o1) * 3 + ch3) * 128 + ys) * 128 + xs] = s;
    }

    const int oA[3] = {0, 2, 1};  // lh->15, hl->75, hh->45
    const int oB[3] = {5, 3, 4};  // lh->165, hl->105, hh->135
#pragma unroll
    for (int q = 0; q < 3; ++q) {
      const float* ob = obuf[1 + q];  // lh, hl, hh
      float a = ob[(2 * u) * 33 + 2 * v] * INV_SQRT2F;
      float bq = ob[(2 * u) * 33 + 2 * v + 1] * INV_SQRT2F;
      float cq = ob[(2 * u + 1) * 33 + 2 * v] * INV_SQRT2F;
      float d = ob[(2 * u + 1) * 33 + 2 * v + 1] * INV_SQRT2F;
      float m1 = smooth_mag(a - d, bq + cq);
      float m2 = smooth_mag(a + d, bq - cq);
      if (MODE == 0) {  // p layout: (b, o*3+ch, 256, 256)
        out_mag[((size_t)(b * 6 * C + oA[q] * C + c_) * Hs + ys) * Hs + xs] = m1;
        out_mag[((size_t)(b * 6 * C + oB[q] * C + c_) * Hs + ys) * Hs + xs] = m2;
      } else {  // s2: Z channel (13 + o2*6 + o1)*3 + ch
        out_mag[((size_t)(b * 147 + (13 + oA[q] * 6 + o1) * 3 + ch3) * 128 + ys) *
                    128 + xs] = m1;
        out_mag[((size_t)(b * 147 + (13 + oB[q] * 6 + o1) * 3 + ch3) * 128 + ys) *
                    128 + xs] = m2;
      }
    }
  }
}

// ---------------------------------------------------------------------------
// Level-2 kernel (fwd_j2plus on ll1 + q2c + mag + avgpool), 10-tap decimated
// filters. One block = (b, ch, 16x16 output tile of the 128x128 grid).
// Row decimation on VALU into parity-split, pair-interleaved LDS planes;
// column decimation on WMMA (dense band after parity split, K=40).
// ---------------------------------------------------------------------------
__global__ __launch_bounds__(256) void scat_level2_kernel(
    const float* __restrict__ LL, const float* __restrict__ g_h0a,
    const float* __restrict__ g_h0b, const float* __restrict__ g_h1a,
    const float* __restrict__ g_h1b, float* __restrict__ Z) {
  // smem layout: shh | aTab2 | pbuf (4 planes) | sx (aliased with obuf)
  __shared__ __align__(16) float smem[40 + 4 * 10 * 64 + 4 * (22 * 66) + 80 * 81];
  float* shh = smem;                       // h0a | h0b | h1a | h1b
  float* aTab2 = smem + 40;                // [hidx][s][lane]{x,y}
  float* pbufm = aTab2 + 4 * 10 * 64;      // loE, loO, hiE, hiO pair-interleaved
  float* sx = pbufm + 4 * (22 * 66);       // 80x81 padded input tile
  float* obufm = sx;                       // alias: sx dead after phase B

  const int tid = threadIdx.x;
  int id = blockIdx.x;
  const int tx = id & 7;
  const int ty = (id >> 3) & 7;
  id >>= 6;
  const int c_ = id % 3;
  const int b = id / 3;

  const float* __restrict__ Xc = LL + (size_t)(b * 3 + c_) * 512 * 512;

  if (tid < 40)
    shh[tid] = (tid < 10)   ? g_h0a[tid]
               : (tid < 20) ? g_h0b[tid - 10]
               : (tid < 30) ? g_h1a[tid - 20]
                            : g_h1b[tid - 30];
  __syncthreads();  // shh ready for aTab2 setup

  // -- precompute A-band fragments: A[r][k] = h[k - 2r], K = 40 ------------
  for (int idx = tid; idx < 4 * 10 * 32; idx += 256) {
    int hidx = idx / 320;
    int rem = idx - hidx * 320;
    int s = rem >> 5, lane = rem & 31;
    int mrow = lane & 15, khalf = (lane >> 4) << 1;
    int k0 = 4 * s + khalf;
    const float* h = shh + hidx * 10;
    int t0 = k0 - 2 * mrow;
    aTab2[idx * 2 + 0] = band_tap(h, t0, 10);
    aTab2[idx * 2 + 1] = band_tap(h, t0 + 1, 10);
  }

  // -- load symmetric-padded 80x80 input tile ------------------------------
  const int ybase = 64 * ty - 8, xbase = 64 * tx - 8;
  for (int idx = tid; idx < 80 * 80; idx += 256) {
    int r = idx / 80, cx = idx - r * 80;
    int gy = reflect_idx(ybase + r, 512);
    int gx = reflect_idx(xbase + cx, 512);
    sx[r * 81 + cx] = Xc[(size_t)gy * 512 + gx];
  }
  __syncthreads();

  // -- rowdfilt (VALU) -> parity-split, pair-interleaved planes ------------
  // lo col even: ya (even samples, h0b); odd: yb (odd samples, h0a)
  // hi col even: yb (odd samples, h1a); odd: ya (even samples, h1b)
  for (int idx = tid; idx < 44 * 32; idx += 256) {
    int jr = idx >> 5, c = idx & 31;
    float loE = 0.f, loO = 0.f, hiE = 0.f, hiO = 0.f;
    if (jr < 40) {
      int l = c >> 1, q = c & 1;
      const float* re = &sx[(2 * jr) * 81];
      const float* ro = &sx[(2 * jr + 1) * 81];
      const float* hLo = q ? (shh + 0) : (shh + 10);
      const float* hHi = q ? (shh + 30) : (shh + 20);
      int baseLo = 4 * l + q;
      int baseHi = 4 * l + (q ? 0 : 1);
#pragma unroll
      for (int t = 0; t < 10; ++t) {
        loE += re[baseLo + 2 * t] * hLo[t];
        loO += ro[baseLo + 2 * t] * hLo[t];
        hiE += re[baseHi + 2 * t] * hHi[t];
        hiO += ro[baseHi + 2 * t] * hHi[t];
      }
    }
    const int a = (jr >> 1) * 66 + c * 2 + (jr & 1);
    pbufm[0 * (22 * 66) + a] = loE;
    pbufm[1 * (22 * 66) + a] = loO;
    pbufm[2 * (22 * 66) + a] = hiE;
    pbufm[3 * (22 * 66) + a] = hiO;
  }
  __syncthreads();

  // -- coldfilt on WMMA: Out(16x16) = A(16x40 band) * B(40x16) -------------
  const int lane = tid & 31, wave = tid >> 5;
  const int mrow = lane & 15, lhi = lane >> 4;
#pragma unroll
  for (int rep = 0; rep < 2; ++rep) {
    const int it = wave + rep * 8;
    const int f = it >> 2;          // 0:ll 1:lh 2:hl 3:hh
    const int par = (it >> 1) & 1;  // 0: even output rows, 1: odd
    const int j = it & 1;
    const int bsel = ((f & 2) ? 2 : 0) + (((f & 1) ^ par) & 1);
    const int hidx = (f & 1) ? (par ? 3 : 2) : (par ? 0 : 1);
    const float* src = pbufm + bsel * (22 * 66);
    const float* at = aTab2 + hidx * (10 * 64);
    const int bcol = (16 * j + mrow) * 2;
    v8f acc = {0.f, 0.f, 0.f, 0.f, 0.f, 0.f, 0.f, 0.f};
#pragma unroll
    for (int s = 0; s < 10; ++s) {
      v2f a = *(const v2f*)&at[(s * 32 + lane) * 2];
      v2f bv = *(const v2f*)&src[(2 * s + lhi) * 66 + bcol];
      acc = wmma16x16x4(a, bv, acc);
    }
    const int rhalf = lhi << 3;
#pragma unroll
    for (int v = 0; v < 8; ++v) {
      int r16 = v + rhalf;
      obufm[f * (32 * 33) + (2 * r16 + par) * 33 + 16 * j + mrow] = acc[v];
    }
  }
  __syncthreads();

  // -- s0 (avgpool of ll2) and s1_j2 magnitudes straight into Z ------------
  const int u = tid >> 4, v = tid & 15;
  const int ys = 16 * ty + u, xs = 16 * tx + v;
  {
    const float* ll = obufm;  // f = 0
    float s = 0.25f * (ll[(2 * u) * 33 + 2 * v] + ll[(2 * u) * 33 + 2 * v + 1] +
                       ll[(2 * u + 1) * 33 + 2 * v] +
                       ll[(2 * u + 1) * 33 + 2 * v + 1]);
    Z[((size_t)(b * 147 + c_) * 128 + ys) * 128 + xs] = s;
  }
  const int oA[3] = {0, 2, 1};
  const int oB[3] = {5, 3, 4};
#pragma unroll
  for (int q = 0; q < 3; ++q) {
    const float* ob = obufm + (1 + q) * (32 * 33);  // lh, hl, hh
    float a = ob[(2 * u) * 33 + 2 * v] * INV_SQRT2F;
    float bq = ob[(2 * u) * 33 + 2 * v + 1] * INV_SQRT2F;
    float cq = ob[(2 * u + 1) * 33 + 2 * v] * INV_SQRT2F;
    float d = ob[(2 * u + 1) * 33 + 2 * v + 1] * INV_SQRT2F;
    float m1 = smooth_mag(a - d, bq + cq);
    float m2 = smooth_mag(a + d, bq - cq);
    Z[((size_t)(b * 147 + (7 + oA[q]) * 3 + c_) * 128 + ys) * 128 + xs] = m1;
    Z[((size_t)(b * 147 + (7 + oB[q]) * 3 + c_) * 128 + ys) * 128 + xs] = m2;
  }
}

// ---------------------------------------------------------------------------
extern "C" void kernel_launch(void* const* d_in, const int* in_sizes, int n_in,
                              void* d_out, int out_size, void* d_ws,
                              size_t ws_size, hipStream_t stream) {
  (void)in_sizes; (void)n_in; (void)out_size; (void)ws_size;
  const float* x = (const float*)d_in[0];
  const float* h0o = (const float*)d_in[1];
  const float* h1o = (const float*)d_in[2];
  const float* h0a = (const float*)d_in[3];
  const float* h0b = (const float*)d_in[4];
  const float* h1a = (const float*)d_in[5];
  const float* h1b = (const float*)d_in[6];
  float* Z = (float*)d_out;

  // Workspace: ll1 (32,3,512,512) then p (32,18,256,256), both fp32.
  float* ll1 = (float*)d_ws;
  float* p = ll1 + (size_t)32 * 3 * 512 * 512;

  // Level 1: x -> ll1, p (= s1_j1 magnitudes, channel = o*3 + ch)
  scat_level1_kernel<0><<<32 * 3 * 16 * 16, 256, 0, stream>>>(
      x, 3, 512, h0o, h1o, ll1, p);
  // Level 2: ll1 -> Z[k=0] (s0) and Z[k=7..12] (s1_j2)
  scat_level2_kernel<<<32 * 3 * 8 * 8, 256, 0, stream>>>(ll1, h0a, h0b, h1a,
                                                         h1b, Z);
  // Second order: p -> Z[k=1..6] (avgpool p_ll) and Z[k=13..48] (s2)
  scat_level1_kernel<1><<<32 * 18 * 8 * 8, 256, 0, stream>>>(
      p, 18, 256, h0o, h1o, nullptr, Z);
}